// VGAEEncoder_46694884442219
// MI455X (gfx1250) — compile-verified
//
#include <hip/hip_runtime.h>
#include <hip/hip_bf16.h>

// ---------------------------------------------------------------------------
// VGAE encoder (GCNConv x3) for MI455X / gfx1250, wave32.
//   conv1: h   = relu( Ahat @ (x @ W1) + b1 )
//   mu    =       Ahat @ (h @ Wmu) + bmu
//   logvar=       Ahat @ (h @ Wlv) + blv
// Ahat = sym-normalized adjacency with self loops.
// Dense transforms use V_WMMA_F32_16X16X4_F32 (native fp32 matrix op).
// Edge aggregation uses fp32 global atomics (resolve in 192MB L2).
// ---------------------------------------------------------------------------

typedef float v2f __attribute__((ext_vector_type(2)));
typedef float v8f __attribute__((ext_vector_type(8)));

#define C_IN  128
#define C_HID 128
#define C_OUT 64

// ---------------------------------------------------------------- degree ----
__global__ void k_deg_init(float* __restrict__ deg, int N) {
    int i = blockIdx.x * blockDim.x + threadIdx.x;
    if (i < N) deg[i] = 1.0f;   // self loop
}

__global__ void k_deg_edges(const int* __restrict__ ei, float* __restrict__ deg,
                            int E) {
    int e = blockIdx.x * blockDim.x + threadIdx.x;
    if (e < E) {
        int c = ei[E + e];      // target node
        __hip_atomic_fetch_add(&deg[c], 1.0f, __ATOMIC_RELAXED,
                               __HIP_MEMORY_SCOPE_AGENT);
    }
}

__global__ void k_deg_to_dinv(float* __restrict__ deg, int N) {
    int i = blockIdx.x * blockDim.x + threadIdx.x;
    if (i < N) {
        float d = deg[i];
        deg[i] = d > 0.0f ? __frsqrt_rn(d) : 0.0f;
    }
}

// ------------------------------------------------------------------ GEMM ----
// h[N,128] = X[N,128] @ W[128,128]
// 1 wave -> one 16x16 output tile; 8 waves/block cover all 8 column tiles.
__global__ void k_gemm_x_w1(const float* __restrict__ X,
                            const float* __restrict__ W,
                            float* __restrict__ H) {
    const int m0   = blockIdx.x * 16;
    const int wave = threadIdx.x >> 5;
    const int lane = threadIdx.x & 31;
    const int n0   = wave * 16;            // 8 waves * 16 = 128 columns
    const int half = lane >> 4;            // 0: lanes 0-15, 1: lanes 16-31
    const int mrow = m0 + (lane & 15);
    const int ncol = n0 + (lane & 15);
    const int kb   = half << 1;            // K sub-offset per lane half

    v8f acc = {};
    #pragma unroll 4
    for (int k = 0; k < C_IN; k += 4) {
        v2f a = *(const v2f*)(X + (size_t)mrow * C_IN + (k + kb));
        v2f b;
        b.x = W[(size_t)(k + kb)     * C_HID + ncol];
        b.y = W[(size_t)(k + kb + 1) * C_HID + ncol];
        acc = __builtin_amdgcn_wmma_f32_16x16x4_f32(
                  false, a, false, b, (short)0, acc, false, false);
    }
    #pragma unroll
    for (int v = 0; v < 8; ++v) {
        int mm = m0 + v + (half << 3);     // D layout: M = v + 8*half
        H[(size_t)mm * C_HID + ncol] = acc[v];
    }
}

// tmu[N,64] = H[N,128] @ Wmu[128,64];  tlv = H @ Wlv.  Waves 0-3: mu, 4-7: lv.
__global__ void k_gemm_h_w23(const float* __restrict__ H,
                             const float* __restrict__ Wmu,
                             const float* __restrict__ Wlv,
                             float* __restrict__ Tmu,
                             float* __restrict__ Tlv) {
    const int m0   = blockIdx.x * 16;
    const int wave = threadIdx.x >> 5;
    const int lane = threadIdx.x & 31;
    const float* __restrict__ W = (wave < 4) ? Wmu : Wlv;
    float* __restrict__ T       = (wave < 4) ? Tmu : Tlv;
    const int n0   = (wave & 3) * 16;      // 4 tiles cover 64 columns
    const int half = lane >> 4;
    const int mrow = m0 + (lane & 15);
    const int ncol = n0 + (lane & 15);
    const int kb   = half << 1;

    v8f acc = {};
    #pragma unroll 4
    for (int k = 0; k < C_HID; k += 4) {
        v2f a = *(const v2f*)(H + (size_t)mrow * C_HID + (k + kb));
        v2f b;
        b.x = W[(size_t)(k + kb)     * C_OUT + ncol];
        b.y = W[(size_t)(k + kb + 1) * C_OUT + ncol];
        acc = __builtin_amdgcn_wmma_f32_16x16x4_f32(
                  false, a, false, b, (short)0, acc, false, false);
    }
    #pragma unroll
    for (int v = 0; v < 8; ++v) {
        int mm = m0 + v + (half << 3);
        T[(size_t)mm * C_OUT + ncol] = acc[v];
    }
}

// ----------------------------------------------------------- aggregation ----
// agg[i,:] = h[i,:] * dinv[i]^2  (self-loop contribution, also clears buffer)
__global__ void k_selfloop_init_h(const float* __restrict__ H,
                                  const float* __restrict__ dinv,
                                  float* __restrict__ A, int N) {
    int t = blockIdx.x * blockDim.x + threadIdx.x;
    int i = t >> 5;                        // node
    if (i >= N) return;
    int c = (t & 31) << 2;                 // 32 lanes * float4 = 128 ch
    float s = dinv[i]; s *= s;
    float4 hv = *(const float4*)(H + (size_t)i * C_HID + c);
    float4 o  = make_float4(hv.x * s, hv.y * s, hv.z * s, hv.w * s);
    *(float4*)(A + (size_t)i * C_HID + c) = o;
}

// per edge: agg[col,:] += h[row,:] * dinv[row]*dinv[col]   (128 channels)
__global__ void k_scatter_h(const int* __restrict__ ei,
                            const float* __restrict__ dinv,
                            const float* __restrict__ H,
                            float* __restrict__ A, int E) {
    int t = blockIdx.x * blockDim.x + threadIdx.x;
    int e = t >> 5;
    if (e >= E) return;
    int c = (t & 31) << 2;
    int r = ei[e];                         // source
    int d = ei[E + e];                     // target
    float nrm = dinv[r] * dinv[d];
    float4 hv = *(const float4*)(H + (size_t)r * C_HID + c);
    float* dst = A + (size_t)d * C_HID + c;
    __hip_atomic_fetch_add(dst + 0, hv.x * nrm, __ATOMIC_RELAXED, __HIP_MEMORY_SCOPE_AGENT);
    __hip_atomic_fetch_add(dst + 1, hv.y * nrm, __ATOMIC_RELAXED, __HIP_MEMORY_SCOPE_AGENT);
    __hip_atomic_fetch_add(dst + 2, hv.z * nrm, __ATOMIC_RELAXED, __HIP_MEMORY_SCOPE_AGENT);
    __hip_atomic_fetch_add(dst + 3, hv.w * nrm, __ATOMIC_RELAXED, __HIP_MEMORY_SCOPE_AGENT);
}

// A = relu(A + b1), in place
__global__ void k_bias_relu(float* __restrict__ A,
                            const float* __restrict__ b, int N) {
    int t = blockIdx.x * blockDim.x + threadIdx.x;
    int i = t >> 5;
    if (i >= N) return;
    int c = (t & 31) << 2;
    float4 v  = *(const float4*)(A + (size_t)i * C_HID + c);
    float4 bb = *(const float4*)(b + c);
    v.x = fmaxf(v.x + bb.x, 0.0f);
    v.y = fmaxf(v.y + bb.y, 0.0f);
    v.z = fmaxf(v.z + bb.z, 0.0f);
    v.w = fmaxf(v.w + bb.w, 0.0f);
    *(float4*)(A + (size_t)i * C_HID + c) = v;
}

// out_mu[i,:] = tmu[i,:]*dinv[i]^2 + bmu ;  out_lv likewise (64 ch each)
__global__ void k_selfloop_init_out(const float* __restrict__ Tmu,
                                    const float* __restrict__ Tlv,
                                    const float* __restrict__ bmu,
                                    const float* __restrict__ blv,
                                    const float* __restrict__ dinv,
                                    float* __restrict__ Omu,
                                    float* __restrict__ Olv, int N) {
    int t = blockIdx.x * blockDim.x + threadIdx.x;
    int i = t >> 5;
    if (i >= N) return;
    int s = t & 31;
    float sc = dinv[i]; sc *= sc;
    const float* src; const float* bias; float* dst; int c;
    if (s < 16) { src = Tmu; bias = bmu; dst = Omu; c = s << 2; }
    else        { src = Tlv; bias = blv; dst = Olv; c = (s - 16) << 2; }
    float4 v  = *(const float4*)(src + (size_t)i * C_OUT + c);
    float4 bb = *(const float4*)(bias + c);
    float4 o  = make_float4(v.x * sc + bb.x, v.y * sc + bb.y,
                            v.z * sc + bb.z, v.w * sc + bb.w);
    *(float4*)(dst + (size_t)i * C_OUT + c) = o;
}

// per edge: out_mu[col,:] += tmu[row,:]*nrm ; out_lv[col,:] += tlv[row,:]*nrm
__global__ void k_scatter_out(const int* __restrict__ ei,
                              const float* __restrict__ dinv,
                              const float* __restrict__ Tmu,
                              const float* __restrict__ Tlv,
                              float* __restrict__ Omu,
                              float* __restrict__ Olv, int E) {
    int t = blockIdx.x * blockDim.x + threadIdx.x;
    int e = t >> 5;
    if (e >= E) return;
    int s = t & 31;
    int r = ei[e];
    int d = ei[E + e];
    float nrm = dinv[r] * dinv[d];
    const float* src; float* dst; int c;
    if (s < 16) { src = Tmu; dst = Omu; c = s << 2; }
    else        { src = Tlv; dst = Olv; c = (s - 16) << 2; }
    float4 v = *(const float4*)(src + (size_t)r * C_OUT + c);
    float* o = dst + (size_t)d * C_OUT + c;
    __hip_atomic_fetch_add(o + 0, v.x * nrm, __ATOMIC_RELAXED, __HIP_MEMORY_SCOPE_AGENT);
    __hip_atomic_fetch_add(o + 1, v.y * nrm, __ATOMIC_RELAXED, __HIP_MEMORY_SCOPE_AGENT);
    __hip_atomic_fetch_add(o + 2, v.z * nrm, __ATOMIC_RELAXED, __HIP_MEMORY_SCOPE_AGENT);
    __hip_atomic_fetch_add(o + 3, v.w * nrm, __ATOMIC_RELAXED, __HIP_MEMORY_SCOPE_AGENT);
}

// ---------------------------------------------------------------- launch ----
extern "C" void kernel_launch(void* const* d_in, const int* in_sizes, int n_in,
                              void* d_out, int out_size, void* d_ws, size_t ws_size,
                              hipStream_t stream) {
    const float* x   = (const float*)d_in[0];
    const int*   ei  = (const int*)  d_in[1];
    const float* W1  = (const float*)d_in[2];
    const float* b1  = (const float*)d_in[3];
    const float* Wmu = (const float*)d_in[4];
    const float* bmu = (const float*)d_in[5];
    const float* Wlv = (const float*)d_in[6];
    const float* blv = (const float*)d_in[7];

    const int N = in_sizes[0] / C_IN;   // 50000
    const int E = in_sizes[1] / 2;      // 800000

    float* ws   = (float*)d_ws;
    float* dinv = ws;                               // N
    float* h    = dinv + N;                         // N*128
    float* agg1 = h    + (size_t)N * C_HID;         // N*128
    float* tmu  = agg1 + (size_t)N * C_HID;         // N*64
    float* tlv  = tmu  + (size_t)N * C_OUT;         // N*64

    float* outmu = (float*)d_out;                   // N*64
    float* outlv = outmu + (size_t)N * C_OUT;       // N*64

    const int T = 256;
    // 1) symmetric normalization with self loops
    k_deg_init   <<<(N + T - 1) / T, T, 0, stream>>>(dinv, N);
    k_deg_edges  <<<(E + T - 1) / T, T, 0, stream>>>(ei, dinv, E);
    k_deg_to_dinv<<<(N + T - 1) / T, T, 0, stream>>>(dinv, N);

    // 2) h = x @ W1   (WMMA fp32)
    k_gemm_x_w1<<<N / 16, T, 0, stream>>>(x, W1, h);

    // 3) conv1 propagate + bias + relu  (result in agg1)
    k_selfloop_init_h<<<(N * 32 + T - 1) / T, T, 0, stream>>>(h, dinv, agg1, N);
    k_scatter_h      <<<(E * 32 + T - 1) / T, T, 0, stream>>>(ei, dinv, h, agg1, E);
    k_bias_relu      <<<(N * 32 + T - 1) / T, T, 0, stream>>>(agg1, b1, N);

    // 4) tmu = agg1 @ Wmu ; tlv = agg1 @ Wlv  (fused, WMMA fp32)
    k_gemm_h_w23<<<N / 16, T, 0, stream>>>(agg1, Wmu, Wlv, tmu, tlv);

    // 5) conv_mu / conv_lv propagate straight into d_out
    k_selfloop_init_out<<<(N * 32 + T - 1) / T, T, 0, stream>>>(
        tmu, tlv, bmu, blv, dinv, outmu, outlv, N);
    k_scatter_out<<<(E * 32 + T - 1) / T, T, 0, stream>>>(
        ei, dinv, tmu, tlv, outmu, outlv, E);
}